// Encoder_57655640982021
// MI455X (gfx1250) — compile-verified
//
#include <hip/hip_runtime.h>
#include <cstdint>
#include <cstddef>

// ---------------------------------------------------------------------------
// Multi-level dense-grid bilinear feature encoder for MI455X (gfx1250).
//
// Roofline: ~0.7 GFLOP total vs ~89 MB of tables -> pure L2-gather problem
// (tables fit in the 192 MB L2). Strategy:
//   1. Re-pack tables (NC,NF,r,r) -> (NC,r,r,NF) in workspace so one bilinear
//      tap = 2x global_load_b128 instead of 8 scattered b32 loads. The repack
//      uses the CDNA5 async global->LDS path (global_load_async_to_lds_b32 +
//      s_wait_asynccnt), with th:TH_LOAD_NT so the one-shot channel-major
//      reads don't pollute L2.
//   2. Gather kernel: 1 thread/point, compile-time resolutions, uniform
//      SGPR-base + u32 per-lane offsets (GVS addressing, minimal VALU),
//      packed-vector FMAs, RT (cached) table gathers, non-temporal output
//      stores and x/y loads so the 160 MB output stream cannot evict the
//      resident tables.
// ---------------------------------------------------------------------------

#define NLEVELS 5
#define NFEAT 8
#define NCELLS 2

typedef float v4f __attribute__((ext_vector_type(4)));

// ---------------------------------------------------------------------------
// Layout transform: in[(cell*NF + c)*R*R + y*R + x] -> out[(cell*R*R + y*R + x)*NF + c]
// Each thread owns one texel. The 8 strided channel reads are issued as async
// non-temporal loads scattered directly into this wave's LDS slice; after
// s_wait_asynccnt 0 the thread reads back its contiguous 8 floats and stores
// two b128s (fully coalesced across lanes, default RT so they stay in L2).
// ---------------------------------------------------------------------------
template <int R>
__global__ void __launch_bounds__(256) transpose_kernel(const float* __restrict__ in,
                                                        float* __restrict__ outT) {
    __shared__ float smem[256 * NFEAT];  // 8 KB
    const int tid = threadIdx.x;
    const size_t t = (size_t)blockIdx.x * 256 + tid;   // texel id: cell*R*R + y*R + x
    const size_t plane = (size_t)R * R;
    const size_t cell = t / plane;
    const size_t yx = t - cell * plane;

    const float* gbase = in + cell * NFEAT * plane + yx;
    // Low 32 bits of a flat shared-space address are the wave-relative LDS
    // byte offset (aperture lives in the high 32 bits).
    const uint32_t lds_base = (uint32_t)(uintptr_t)(&smem[tid * NFEAT]);

#pragma unroll
    for (int c = 0; c < NFEAT; ++c) {
        const float* gp = gbase + (size_t)c * plane;
        uint32_t la = lds_base + (uint32_t)(c * 4);
        asm volatile("global_load_async_to_lds_b32 %0, %1, off th:TH_LOAD_NT"
                     :
                     : "v"(la), "v"(gp)
                     : "memory");
    }
    asm volatile("s_wait_asynccnt 0x0" ::: "memory");

    const v4f lo = *(const v4f*)(&smem[tid * NFEAT + 0]);
    const v4f hi = *(const v4f*)(&smem[tid * NFEAT + 4]);
    v4f* op = (v4f*)(outT + t * NFEAT);
    op[0] = lo;   // RT: keep transposed tables resident in L2
    op[1] = hi;
}

// ---------------------------------------------------------------------------
// One level, one point: bilinear sample with per-cell staggered offsets,
// summed over the 2 cells. TR selects texel-major (fast) vs original layout.
// All table indexing uses unsigned 32-bit element offsets from a uniform base
// (largest table = 16.8M floats -> fits u32 byte offsets), so the backend can
// use SADDR + 32-bit VGPR-offset global addressing instead of per-lane 64-bit
// pointer arithmetic.
// ---------------------------------------------------------------------------
template <int W, bool TR>
__device__ __forceinline__ void sample_level(const float* __restrict__ tab,
                                             float xn, float yn,
                                             v4f& alo, v4f& ahi) {
#pragma unroll
    for (int cell = 0; cell < NCELLS; ++cell) {
        const float off = (float)cell * 0.5f;
        const float ix = (xn + 1.0f) * 0.5f * (float)(W - 1) + off;
        const float iy = (yn + 1.0f) * 0.5f * (float)(W - 1) + off;
        const float x0f = floorf(ix);
        const float y0f = floorf(iy);
        const float wx = ix - x0f;
        const float wy = iy - y0f;
        int x0 = (int)x0f;
        x0 = x0 < 0 ? 0 : (x0 > W - 1 ? W - 1 : x0);
        int y0 = (int)y0f;
        y0 = y0 < 0 ? 0 : (y0 > W - 1 ? W - 1 : y0);
        const int x1 = (x0 + 1 > W - 1) ? (W - 1) : (x0 + 1);
        const int y1 = (y0 + 1 > W - 1) ? (W - 1) : (y0 + 1);

        const float w00 = (1.0f - wx) * (1.0f - wy);
        const float w01 = wx * (1.0f - wy);
        const float w10 = (1.0f - wx) * wy;
        const float w11 = wx * wy;

        const unsigned r0 = (unsigned)(y0 * W);
        const unsigned r1 = (unsigned)(y1 * W);

        if (TR) {
            const float* b = tab + (unsigned)(cell * W * W * NFEAT);  // uniform base
            const unsigned i00 = (r0 + (unsigned)x0) * NFEAT;
            const unsigned i01 = (r0 + (unsigned)x1) * NFEAT;
            const unsigned i10 = (r1 + (unsigned)x0) * NFEAT;
            const unsigned i11 = (r1 + (unsigned)x1) * NFEAT;
            const unsigned idx[4] = {i00, i01, i10, i11};
            const float wgt[4] = {w00, w01, w10, w11};
#pragma unroll
            for (int k = 0; k < 4; ++k) {
                const v4f lo = *(const v4f*)(b + idx[k]);       // global_load_b128 (RT)
                const v4f hi = *(const v4f*)(b + idx[k] + 4u);
                const float w = wgt[k];
                alo += w * lo;                                  // packed vector FMAs
                ahi += w * hi;
            }
        } else {
            const float* b = tab + (unsigned)(cell * NFEAT * W * W);  // uniform base
            const unsigned i00 = r0 + (unsigned)x0;
            const unsigned i01 = r0 + (unsigned)x1;
            const unsigned i10 = r1 + (unsigned)x0;
            const unsigned i11 = r1 + (unsigned)x1;
#pragma unroll
            for (int c = 0; c < 4; ++c) {
                const float* pl = b + (unsigned)(c * W * W);
                float a = alo[c];
                a = fmaf(w00, pl[i00], a);
                a = fmaf(w01, pl[i01], a);
                a = fmaf(w10, pl[i10], a);
                a = fmaf(w11, pl[i11], a);
                alo[c] = a;
            }
#pragma unroll
            for (int c = 0; c < 4; ++c) {
                const float* pl = b + (unsigned)((c + 4) * W * W);
                float a = ahi[c];
                a = fmaf(w00, pl[i00], a);
                a = fmaf(w01, pl[i01], a);
                a = fmaf(w10, pl[i10], a);
                a = fmaf(w11, pl[i11], a);
                ahi[c] = a;
            }
        }
    }
}

template <bool TR>
__global__ void __launch_bounds__(256) encoder_kernel(const float* __restrict__ xin,
                                                      const float* __restrict__ yin,
                                                      const float* __restrict__ t0,
                                                      const float* __restrict__ t1,
                                                      const float* __restrict__ t2,
                                                      const float* __restrict__ t3,
                                                      const float* __restrict__ t4,
                                                      float* __restrict__ out, int P) {
    const int p = blockIdx.x * blockDim.x + threadIdx.x;
    if (p >= P) return;

    // x/y are streamed once: non-temporal loads, don't displace table lines.
    const float xn = __builtin_nontemporal_load(&xin[p]) * 2.0f - 1.0f;
    const float yn = __builtin_nontemporal_load(&yin[p]) * 2.0f - 1.0f;
    // 160 B rows, 16 B aligned; single widening multiply for the row base.
    v4f* o = (v4f*)(out + (size_t)((unsigned)p) * (NLEVELS * NFEAT));

#define DO_LEVEL(W, TAB, IDX)                                       \
    {                                                               \
        v4f alo = {0.f, 0.f, 0.f, 0.f};                             \
        v4f ahi = {0.f, 0.f, 0.f, 0.f};                             \
        sample_level<W, TR>(TAB, xn, yn, alo, ahi);                 \
        __builtin_nontemporal_store(alo, &o[(IDX) * 2 + 0]);        \
        __builtin_nontemporal_store(ahi, &o[(IDX) * 2 + 1]);        \
    }

    DO_LEVEL(64, t0, 0)
    DO_LEVEL(128, t1, 1)
    DO_LEVEL(256, t2, 2)
    DO_LEVEL(512, t3, 3)
    DO_LEVEL(1024, t4, 4)
#undef DO_LEVEL
}

// ---------------------------------------------------------------------------
// Host launcher
// ---------------------------------------------------------------------------
extern "C" void kernel_launch(void* const* d_in, const int* in_sizes, int n_in,
                              void* d_out, int out_size, void* d_ws, size_t ws_size,
                              hipStream_t stream) {
    const float* x = (const float*)d_in[0];
    const float* y = (const float*)d_in[1];
    const float* tabs[NLEVELS] = {
        (const float*)d_in[2], (const float*)d_in[3], (const float*)d_in[4],
        (const float*)d_in[5], (const float*)d_in[6]};

    float* out = (float*)d_out;
    const int P = in_sizes[0];

    static const int RESH[NLEVELS] = {64, 128, 256, 512, 1024};
    size_t offs[NLEVELS];
    size_t need = 0;
    for (int i = 0; i < NLEVELS; ++i) {
        offs[i] = need;  // in floats
        need += (size_t)NCELLS * RESH[i] * RESH[i] * NFEAT;
    }
    const size_t need_bytes = need * sizeof(float);

    const int threads = 256;
    const int grid = (P + threads - 1) / threads;

    if (d_ws != nullptr && ws_size >= need_bytes) {
        float* w = (float*)d_ws;
        transpose_kernel<64><<<(2 * 64 * 64) / 256, 256, 0, stream>>>(tabs[0], w + offs[0]);
        transpose_kernel<128><<<(2 * 128 * 128) / 256, 256, 0, stream>>>(tabs[1], w + offs[1]);
        transpose_kernel<256><<<(2 * 256 * 256) / 256, 256, 0, stream>>>(tabs[2], w + offs[2]);
        transpose_kernel<512><<<(2 * 512 * 512) / 256, 256, 0, stream>>>(tabs[3], w + offs[3]);
        transpose_kernel<1024><<<(2 * 1024 * 1024) / 256, 256, 0, stream>>>(tabs[4], w + offs[4]);
        encoder_kernel<true><<<grid, threads, 0, stream>>>(
            x, y, w + offs[0], w + offs[1], w + offs[2], w + offs[3], w + offs[4], out, P);
    } else {
        encoder_kernel<false><<<grid, threads, 0, stream>>>(
            x, y, tabs[0], tabs[1], tabs[2], tabs[3], tabs[4], out, P);
    }
}